// ColorHistograms_71794673320474
// MI455X (gfx1250) — compile-verified
//
#include <hip/hip_runtime.h>
#include <hip/hip_bf16.h>

typedef __attribute__((ext_vector_type(2))) float v2f;
typedef __attribute__((ext_vector_type(8))) float v8f;

#define T_DIM   1024
#define WIN     101
#define K_PAD   104       // WIN padded to multiple of 4 for 16x16x4 WMMA
#define PADW    50
#define NOUT    128
#define HWC     3888      // 27*48*3 contiguous floats per (b,t)
#define NPIX    1296      // 27*48

// ---------------------------------------------------------------------------
// k0: transpose fc_w (128 x 101, row-major) into wT[K_PAD][NOUT], zero-padded
// ---------------------------------------------------------------------------
__global__ __launch_bounds__(128) void fcw_transpose_kernel(
    const float* __restrict__ fc_w, float* __restrict__ wT) {
  const int k = blockIdx.x;       // 0..103
  const int o = threadIdx.x;      // 0..127
  float v = 0.0f;
  if (k < WIN) v = fc_w[o * WIN + k];
  wT[k * NOUT + o] = v;
}

// ---------------------------------------------------------------------------
// k1: per-(b,t) spatial mean over 27*48 pixels, 3 channels.
// One 128-thread block per (b,t); the 3888 floats are contiguous in memory.
// ---------------------------------------------------------------------------
__global__ __launch_bounds__(128) void mean_colors_kernel(
    const float* __restrict__ x, float* __restrict__ means) {
  const int bt = blockIdx.x;                       // 0..16383
  const float* __restrict__ p = x + (size_t)bt * HWC;

  float s0 = 0.f, s1 = 0.f, s2 = 0.f;
  for (int pix = threadIdx.x; pix < NPIX; pix += 128) {
    const float* q = p + pix * 3;                  // 12B per thread, contiguous
    s0 += q[0]; s1 += q[1]; s2 += q[2];            // across the wave
  }
  // wave32 reduction
  #pragma unroll
  for (int off = 16; off > 0; off >>= 1) {
    s0 += __shfl_down(s0, off, 32);
    s1 += __shfl_down(s1, off, 32);
    s2 += __shfl_down(s2, off, 32);
  }
  __shared__ float sw[4][3];
  const int lane = threadIdx.x & 31;
  const int wv   = threadIdx.x >> 5;
  if (lane == 0) { sw[wv][0] = s0; sw[wv][1] = s1; sw[wv][2] = s2; }
  __syncthreads();
  if (threadIdx.x == 0) {
    const float inv = 1.0f / (float)NPIX;
    float r0 = sw[0][0] + sw[1][0] + sw[2][0] + sw[3][0];
    float r1 = sw[0][1] + sw[1][1] + sw[2][1] + sw[3][1];
    float r2 = sw[0][2] + sw[1][2] + sw[2][2] + sw[3][2];
    means[bt * 3 + 0] = r0 * inv;
    means[bt * 3 + 1] = r1 * inv;
    means[bt * 3 + 2] = r2 * inv;
  }
}

// ---------------------------------------------------------------------------
// k2: window features + fp32 WMMA GEMM + bias + ReLU.
// Block = 256 threads (8 waves) handles one 16-row M-tile x full N=128.
// Each wave owns one 16-wide N-tile: 26 x v_wmma_f32_16x16x4_f32.
// ---------------------------------------------------------------------------
__global__ __launch_bounds__(256) void window_gemm_kernel(
    const float* __restrict__ means, const float* __restrict__ wT,
    const float* __restrict__ fc_b, float* __restrict__ out) {
  __shared__ float s_meanj[116 * 3];       // means for j in [t0-50, t0+65]
  __shared__ float s_win[16][106];         // stride 106: conflict-free, 8B-aligned

  const int blk = blockIdx.x;              // 0..1023
  const int b   = blk >> 6;                // blk / 64
  const int t0  = (blk & 63) << 4;         // tile start within T
  const int tid = threadIdx.x;

  // Stage the 116 mean rows this tile can touch (zeros for out-of-range j).
  for (int i = tid; i < 116 * 3; i += 256) {
    const int jj = i / 3, c = i - jj * 3;
    const int j  = t0 - PADW + jj;
    float v = 0.0f;
    if (j >= 0 && j < T_DIM) v = means[((size_t)b * T_DIM + j) * 3 + c];
    s_meanj[i] = v;
  }
  __syncthreads();

  // Materialize 16 x 104 window tile (cols 101..103 zero).
  for (int i = tid; i < 16 * K_PAD; i += 256) {
    const int m = i / K_PAD, w = i - m * K_PAD;
    float v = 0.0f;
    if (w < WIN) {
      const int j = t0 + m + w - PADW;
      if (j >= 0 && j < T_DIM) {
        const float* mr = &s_meanj[(PADW + m) * 3];   // row t0+m
        const float* mj = &s_meanj[(m + w) * 3];      // row j
        v = fabsf(mr[0] - mj[0]) + fabsf(mr[1] - mj[1]) + fabsf(mr[2] - mj[2]);
      }
    }
    s_win[m][w] = v;
  }
  __syncthreads();

  const int lane  = tid & 31;
  const int wave  = tid >> 5;              // 0..7 -> N tile
  const int nbase = wave * 16;
  const int mrow  = lane & 15;             // A/C column mapping
  const int khalf = (lane >> 4) * 2;       // lanes 16-31 hold K+2,K+3
  const int n     = nbase + mrow;

  v8f acc = {};
  #pragma unroll 4
  for (int k0 = 0; k0 < K_PAD; k0 += 4) {
    v2f a, bf;
    // A (16x4 f32): VGPR0 = K{k0, k0+2}, VGPR1 = K{k0+1, k0+3} per half-wave
    a.x = s_win[mrow][k0 + khalf];
    a.y = s_win[mrow][k0 + khalf + 1];
    // B (4x16 f32): row K striped across 16 lanes, mirrored half-wave split
    bf.x = wT[(k0 + khalf) * NOUT + n];
    bf.y = wT[(k0 + khalf + 1) * NOUT + n];
    acc = __builtin_amdgcn_wmma_f32_16x16x4_f32(
        /*neg_a=*/false, a, /*neg_b=*/false, bf,
        /*c_mod=*/(short)0, acc, /*reuse_a=*/false, /*reuse_b=*/false);
  }

  // Epilogue: C/D layout -> lanes 0-15: M=i, lanes 16-31: M=8+i; fuse bias+ReLU.
  const float bias  = fc_b[n];
  const size_t mb   = (size_t)b * T_DIM + t0 + ((lane >> 4) << 3);
  #pragma unroll
  for (int i = 0; i < 8; ++i) {
    float v = acc[i] + bias;
    out[(mb + i) * NOUT + n] = v > 0.0f ? v : 0.0f;
  }
}

// ---------------------------------------------------------------------------
extern "C" void kernel_launch(void* const* d_in, const int* in_sizes, int n_in,
                              void* d_out, int out_size, void* d_ws, size_t ws_size,
                              hipStream_t stream) {
  const float* x    = (const float*)d_in[0];   // (16,1024,27,48,3) f32
  const float* fc_w = (const float*)d_in[1];   // (128,101) f32
  const float* fc_b = (const float*)d_in[2];   // (128,) f32
  float* out = (float*)d_out;                  // (16,1024,128) f32

  float* means = (float*)d_ws;                 // 16384*3 floats
  float* wT    = means + 16384 * 3;            // 104*128 floats

  fcw_transpose_kernel<<<K_PAD, 128, 0, stream>>>(fc_w, wT);
  mean_colors_kernel<<<16 * T_DIM, 128, 0, stream>>>(x, means);
  window_gemm_kernel<<<(16 * T_DIM) / 16, 256, 0, stream>>>(means, wT, fc_b, out);
}